// ZeroWeave_89601607729830
// MI455X (gfx1250) — compile-verified
//
#include <hip/hip_runtime.h>
#include <stdint.h>

// ZeroWeave: out[:, :, ::3, ::3] = x, zeros elsewhere.
// x: [32, 64, 64, 64] f32  ->  out: [32, 64, 190, 190] f32
//
// Pure data movement (0 FLOPs): 33 MiB read + 282 MiB written -> ~13us floor
// @ 23.3 TB/s, store-bound. Round-1 version was instruction-issue-bound
// (~680 insts / 64 B / thread); this version assigns one wave per 3-row
// group so the interleave pattern is uniform and cheap:
//   - rows are exactly 95 float2 and always 8B-aligned -> b64 stores only,
//     no alignment cases anywhere
//   - data row: period-3 float2 pattern {x,0},{0,x},{0,0}; x col = (2g+1)/3
//   - the two zero rows (66% of bytes) stream through the gfx1250 async
//     LDS->global store path (ASYNCcnt), sourced from a zeroed LDS tile.

typedef float v2f __attribute__((ext_vector_type(2)));

constexpr unsigned IMG_N    = 32u * 64u;   // 2048 images (B*C)
constexpr unsigned HW       = 64u;         // input H = W
constexpr unsigned OUT_HW   = 190u;        // output H = W
constexpr unsigned ROW_F2   = OUT_HW / 2u; // 95 float2 per output row
constexpr unsigned GRP_F2   = 3u * ROW_F2; // 285 float2 per 3-row group
constexpr unsigned IMG_F2   = OUT_HW * ROW_F2; // 18050 float2 per image
constexpr unsigned JOBS_PER_IMG = 64u;     // 63 full groups + final data row
constexpr unsigned BLOCK    = 256u;        // 8 waves
constexpr unsigned WAVES_PER_BLOCK = BLOCK / 32u;
constexpr unsigned TOTAL_WAVES = IMG_N * JOBS_PER_IMG;          // 131072
constexpr unsigned GRID     = TOTAL_WAVES / WAVES_PER_BLOCK;    // 16384

__global__ __launch_bounds__(BLOCK)
void zeroweave_kernel(const float* __restrict__ x, float* __restrict__ out) {
    // 16 bytes of zeros in LDS: source tile for async zero stores.
    __shared__ v2f zero_tile[2];
    if (threadIdx.x == 0) {
        zero_tile[0] = (v2f){0.f, 0.f};
        zero_tile[1] = (v2f){0.f, 0.f};
    }
    __syncthreads();
    // Generic pointers to LDS are {shared_aperture_hi, lds_byte_offset}:
    // low 32 bits == 0-based LDS byte address for the async-store VSRC.
    unsigned zoff = (unsigned)(unsigned long long)(uintptr_t)&zero_tile[0];

    const unsigned lane = threadIdx.x & 31u;
    const unsigned wid  = blockIdx.x * WAVES_PER_BLOCK + (threadIdx.x >> 5);
    const unsigned img  = wid >> 6;       // / 64
    const unsigned j    = wid & 63u;      // job within image

    // Input row for this job's data row (row 3j, or 189 when j==63 -> hr=63=j).
    const float* __restrict__ xrow = x + img * (HW * HW) + j * HW;
    // Output base (float2 units): img*18050 + j*285 covers both job kinds.
    v2f* __restrict__ orow = (v2f*)out + (img * IMG_F2 + j * GRP_F2);

    // ---- Phase 1: the data row (95 float2, pattern period 3) ----
    #pragma unroll
    for (int it = 0; it < 3; ++it) {
        unsigned g = lane + 32u * it;
        if (g < ROW_F2) {
            unsigned p    = g % 3u;              // 0:{x,0} 1:{0,x} 2:{0,0}
            unsigned wcol = (2u * g + 1u) / 3u;  // valid x col for p==0 and p==1
            float xv = xrow[wcol];               // wcol <= 63, always in range
            v2f v;
            v.x = (p == 0u) ? xv : 0.0f;
            v.y = (p == 1u) ? xv : 0.0f;
            __builtin_nontemporal_store(v, orow + g);
        }
    }

    // ---- Phase 2: two all-zero rows (190 float2), full groups only ----
    if (j < 63u) {
        #pragma unroll
        for (int it = 0; it < 6; ++it) {
            unsigned g = ROW_F2 + lane + 32u * it;
            if (g < GRP_F2) {
                unsigned long long ga = (unsigned long long)(uintptr_t)(orow + g);
                // 8B of zeros, LDS -> global, tracked by ASYNCcnt.
                asm volatile("global_store_async_from_lds_b64 %0, %1, off"
                             :: "v"(ga), "v"(zoff)
                             : "memory");
            }
        }
    }

    // Drain outstanding async LDS->global stores before wave exit.
    asm volatile("s_wait_asynccnt 0" ::: "memory");
}

extern "C" void kernel_launch(void* const* d_in, const int* in_sizes, int n_in,
                              void* d_out, int out_size, void* d_ws, size_t ws_size,
                              hipStream_t stream) {
    (void)in_sizes; (void)n_in; (void)d_ws; (void)ws_size; (void)out_size;
    const float* x = (const float*)d_in[0];
    // d_in[1] is num_zeros == 2 (static shapes); kernel specialized for s=3.
    float* out = (float*)d_out;
    zeroweave_kernel<<<dim3(GRID), dim3(BLOCK), 0, stream>>>(x, out);
}